// force_profile_6983616824409
// MI455X (gfx1250) — compile-verified
//
#include <hip/hip_runtime.h>

// ---------------------------------------------------------------------------
// Double exclusive prefix sum (moment of a force profile) for P = 2^25 fp32.
// 3-pass tile scan:
//   pass1: per-8192 tile aggregates (B = sum f, D = sum (L-1-j) f)  [WMMA reduce]
//   pass2: single-block scan of 4096 tile triples -> per-tile carry (E, M)
//   pass3: local double-scan per tile + carry; global<->LDS staged with the
//          gfx1250 async data movers (ASYNCcnt) when available.
// Memory-bound: ~384 MB traffic @ 23.3 TB/s  =>  ~16 us ceiling.
// ---------------------------------------------------------------------------

typedef float v2f __attribute__((ext_vector_type(2)));
typedef float v8f __attribute__((ext_vector_type(8)));

#define TILE   8192
#define BDIM   256
#define ITEMS  32      // elements per thread in pass3 (BDIM*ITEMS == TILE)
#define P2_G   16      // aggregates per thread in pass2 (BDIM*P2_G == 4096)
#define NWAVE  (BDIM / 32)

// gfx1250 async global<->LDS movers (guarded so absence still compiles).
#if defined(__has_builtin)
#if __has_builtin(__builtin_amdgcn_global_load_async_to_lds_b128) && \
    __has_builtin(__builtin_amdgcn_global_store_async_from_lds_b128) && \
    __has_builtin(__builtin_amdgcn_s_wait_asynccnt)
#define HAVE_ASYNC_LDS 1
#endif
#endif
#ifndef HAVE_ASYNC_LDS
#define HAVE_ASYNC_LDS 0
#endif

// LDS skew: +16B per 32-float row. Keeps every 16B-aligned 4-float chunk both
// contiguous and 16B-aligned (required for B128 async movers); stride-32
// compute reads see at most a 2-way bank conflict (36t+j mod 64).
__device__ __forceinline__ int lpad(int i) { return i + 4 * (i >> 5); }
#define TILE_PAD (TILE + 4 * (TILE / 32))

struct Trip { float B, D, L; };

// (B1,D1,L1) then (B2,D2,L2): D = D1 + D2 + L2*B1 (associative).
__device__ __forceinline__ Trip trip_combine(const Trip& a, const Trip& b) {
  Trip r;
  r.B = a.B + b.B;
  r.D = a.D + b.D + b.L * a.B;
  r.L = a.L + b.L;
  return r;
}

// ---------------------------------------------------------------------------
// Wave(32)-wide sum via V_WMMA_F32_16X16X4_F32.
// A fragment (documented 16x4 f32 layout) carries x in K-slot 0, 0 in K-slot 1;
// B = all-ones (layout-invariant), C = 0. D[m][n] = x_m + x_{m+16} replicated
// across columns, so sum of the 8 D vgprs + shfl_xor(16) = full wave sum.
// EXEC is all-ones at every call site (uniform control flow, full blocks).
// ---------------------------------------------------------------------------
__device__ __forceinline__ float wave_sum_wmma(float x) {
  v2f a;     a[0] = x;       a[1] = 0.0f;
  v2f bone;  bone[0] = 1.0f; bone[1] = 1.0f;
  v8f c = {0.f, 0.f, 0.f, 0.f, 0.f, 0.f, 0.f, 0.f};
  v8f d = __builtin_amdgcn_wmma_f32_16x16x4_f32(false, a, false, bone,
                                                (short)0, c, false, false);
  float s = d[0] + d[1] + d[2] + d[3] + d[4] + d[5] + d[6] + d[7];
  s += __shfl_xor(s, 16, 32);
  return s;
}

__device__ __forceinline__ Trip wave_inclusive_scan(Trip t, int lane) {
  #pragma unroll
  for (int d = 1; d < 32; d <<= 1) {
    Trip o;
    o.B = __shfl_up(t.B, d, 32);
    o.D = __shfl_up(t.D, d, 32);
    o.L = __shfl_up(t.L, d, 32);
    if (lane >= d) t = trip_combine(o, t);
  }
  return t;
}

// Exclusive triple scan across a 256-thread block (8 wave32 waves).
__device__ __forceinline__ Trip block_exclusive_scan(Trip t, int tid,
                                                     Trip* sWave /*[NWAVE]*/) {
  const int lane = tid & 31, wave = tid >> 5;
  Trip inc = wave_inclusive_scan(t, lane);
  if (lane == 31) sWave[wave] = inc;
  __syncthreads();
  if (tid == 0) {
    Trip run; run.B = 0.f; run.D = 0.f; run.L = 0.f;
    #pragma unroll
    for (int w = 0; w < NWAVE; ++w) {
      Trip tmp = sWave[w];
      sWave[w] = run;                 // exclusive wave prefix
      run = trip_combine(run, tmp);
    }
  }
  __syncthreads();
  Trip wpre = sWave[wave];
  Trip ex;
  ex.B = __shfl_up(inc.B, 1, 32);
  ex.D = __shfl_up(inc.D, 1, 32);
  ex.L = __shfl_up(inc.L, 1, 32);
  if (lane == 0) { ex.B = 0.f; ex.D = 0.f; ex.L = 0.f; }
  return trip_combine(wpre, ex);
}

// ---------------------------------------------------------------------------
// Pass 1: per-tile (B, D). Coalesced b128 loads; WMMA wave reductions.
// ---------------------------------------------------------------------------
__global__ void __launch_bounds__(BDIM) pass1_aggregate(
    const float* __restrict__ f, float2* __restrict__ agg) {
  const int tid = threadIdx.x;
  const long long base = (long long)blockIdx.x * TILE;
  const float4* f4 = (const float4*)(f + base);
  float b = 0.f, d = 0.f;
  #pragma unroll
  for (int k = 0; k < TILE / (4 * BDIM); ++k) {         // 8 iterations
    const int i  = tid + k * BDIM;
    const float4 v = f4[i];
    const int j0 = 4 * i;
    b += v.x + v.y + v.z + v.w;
    d += (float)(TILE - 1 - j0) * v.x + (float)(TILE - 2 - j0) * v.y
       + (float)(TILE - 3 - j0) * v.z + (float)(TILE - 4 - j0) * v.w;
  }
  const float wb = wave_sum_wmma(b);
  const float wd = wave_sum_wmma(d);
  __shared__ float sb[NWAVE], sd[NWAVE];
  const int lane = tid & 31, wave = tid >> 5;
  if (lane == 0) { sb[wave] = wb; sd[wave] = wd; }
  __syncthreads();
  if (tid == 0) {
    float B = 0.f, D = 0.f;
    #pragma unroll
    for (int w = 0; w < NWAVE; ++w) { B += sb[w]; D += sd[w]; }
    agg[blockIdx.x] = make_float2(B, D);
  }
}

// ---------------------------------------------------------------------------
// Pass 2: single block scans all tile triples -> per-tile carry (E = S1, M = S2).
// ---------------------------------------------------------------------------
__global__ void __launch_bounds__(BDIM) pass2_scan(
    const float2* __restrict__ agg, float2* __restrict__ carry, int nblk) {
  __shared__ Trip sWave[NWAVE];
  const int tid = threadIdx.x;
  float2 loc[P2_G];
  Trip t; t.B = 0.f; t.D = 0.f; t.L = 0.f;
  #pragma unroll
  for (int i = 0; i < P2_G; ++i) {
    const int idx = tid * P2_G + i;
    const float2 a = (idx < nblk) ? agg[idx] : make_float2(0.f, 0.f);
    loc[i] = a;
    Trip e; e.B = a.x; e.D = a.y; e.L = (float)TILE;
    t = trip_combine(t, e);
  }
  const Trip ex = block_exclusive_scan(t, tid, sWave);
  float S1 = ex.B;          // applying exclusive segment to (0,0)
  float S2 = ex.D;
  #pragma unroll
  for (int i = 0; i < P2_G; ++i) {
    const int idx = tid * P2_G + i;
    if (idx < nblk) carry[idx] = make_float2(S1, S2);
    S2 += (float)TILE * S1 + loc[i].y;
    S1 += loc[i].x;
  }
}

// ---------------------------------------------------------------------------
// Pass 3: local double-scan per tile + global carry; LDS-staged I/O through
// the async data movers when the toolchain exposes them.
// ---------------------------------------------------------------------------
__global__ void __launch_bounds__(BDIM) pass3_output(
    const float* __restrict__ f, const float* __restrict__ interval,
    const float2* __restrict__ carry, float* __restrict__ out) {
  __shared__ float tile_s[TILE_PAD];
  __shared__ Trip  sWave[NWAVE];
  const int tid = threadIdx.x;
  const long long base = (long long)blockIdx.x * TILE;
  const float4* f4 = (const float4*)(f + base);

#if HAVE_ASYNC_LDS
  // Builtin signature (pinned down by the round-2/3 diagnostics):
  //   (v4i_vec* gaddr, __shared__ v4i_vec* ldsaddr, int offset, int cpol)
  typedef int v4i_vec __attribute__((vector_size(4 * sizeof(int))));
  typedef __attribute__((address_space(3))) v4i_vec lv4i_t;
  #pragma unroll
  for (int k = 0; k < TILE / (4 * BDIM); ++k) {
    const int i  = tid + k * BDIM;
    // LDS offset = low 32 bits of the flat shared address (aperture mapping).
    const unsigned loff = (unsigned)(unsigned long long)&tile_s[lpad(4 * i)];
    __builtin_amdgcn_global_load_async_to_lds_b128(
        (v4i_vec*)(f4 + i), (lv4i_t*)loff, 0, 0);
  }
  __builtin_amdgcn_s_wait_asynccnt(0);
  __syncthreads();
#else
  #pragma unroll
  for (int k = 0; k < TILE / (4 * BDIM); ++k) {
    const int i  = tid + k * BDIM;
    const float4 v = f4[i];
    const int j0 = 4 * i;
    tile_s[lpad(j0 + 0)] = v.x;
    tile_s[lpad(j0 + 1)] = v.y;
    tile_s[lpad(j0 + 2)] = v.z;
    tile_s[lpad(j0 + 3)] = v.w;
  }
  __syncthreads();
#endif

  float vals[ITEMS];
  float b = 0.f, d = 0.f;
  #pragma unroll
  for (int j = 0; j < ITEMS; ++j) {
    const float v = tile_s[lpad(tid * ITEMS + j)];
    vals[j] = v;
    b += v;
    d += (float)(ITEMS - 1 - j) * v;
  }
  Trip t; t.B = b; t.D = d; t.L = (float)ITEMS;
  const Trip ex = block_exclusive_scan(t, tid, sWave);

  const float2 cr = carry[blockIdx.x];
  const float  iv = interval[0];
  float S1 = cr.x + ex.B;                    // E + local exclusive sum
  float S2 = cr.y + ex.L * cr.x + ex.D;      // M + Lex*E + local exclusive moment

  #pragma unroll
  for (int j = 0; j < ITEMS; ++j) {
    tile_s[lpad(tid * ITEMS + j)] = iv * S2;
    S2 += S1;
    S1 += vals[j];
  }
  __syncthreads();

  float4* o4 = (float4*)(out + base);
#if HAVE_ASYNC_LDS
  #pragma unroll
  for (int k = 0; k < TILE / (4 * BDIM); ++k) {
    const int i  = tid + k * BDIM;
    const unsigned loff = (unsigned)(unsigned long long)&tile_s[lpad(4 * i)];
    __builtin_amdgcn_global_store_async_from_lds_b128(
        (v4i_vec*)(o4 + i), (lv4i_t*)loff, 0, 0);
  }
  __builtin_amdgcn_s_wait_asynccnt(0);
#else
  #pragma unroll
  for (int k = 0; k < TILE / (4 * BDIM); ++k) {
    const int i  = tid + k * BDIM;
    const int j0 = 4 * i;
    float4 v;
    v.x = tile_s[lpad(j0 + 0)];
    v.y = tile_s[lpad(j0 + 1)];
    v.z = tile_s[lpad(j0 + 2)];
    v.w = tile_s[lpad(j0 + 3)];
    o4[i] = v;
  }
#endif
}

// ---------------------------------------------------------------------------
extern "C" void kernel_launch(void* const* d_in, const int* in_sizes, int n_in,
                              void* d_out, int out_size, void* d_ws, size_t ws_size,
                              hipStream_t stream) {
  const float* forces   = (const float*)d_in[0];
  // d_in[1] = positions (unused by the reference moment computation)
  const float* interval = (const float*)d_in[2];
  float* out = (float*)d_out;

  const int P    = in_sizes[0];
  const int nblk = P / TILE;            // 4096 for P = 2^25

  float2* agg   = (float2*)d_ws;        // nblk float2
  float2* carry = agg + nblk;           // nblk float2

  pass1_aggregate<<<nblk, BDIM, 0, stream>>>(forces, agg);
  pass2_scan<<<1, BDIM, 0, stream>>>(agg, carry, nblk);
  pass3_output<<<nblk, BDIM, 0, stream>>>(forces, interval, carry, out);
}